// MyLSTM_39599598469282
// MI455X (gfx1250) — compile-verified
//
#include <hip/hip_runtime.h>
#include <hip/hip_bf16.h>

typedef float v2f __attribute__((ext_vector_type(2)));
typedef float v4f __attribute__((ext_vector_type(4)));
typedef float v8f __attribute__((ext_vector_type(8)));

#define B_DIM 8192
#define E_DIM 1024
#define H_DIM 1024
#define K_DIM (E_DIM + H_DIM)   // 2048

#define BM   128     // output rows per workgroup (4 waves x 2 sub-tiles x 16)
#define BN   16      // output cols per workgroup (per gate)
#define KB   32      // K-block staged in LDS (double-buffered)
#define NKB  (K_DIM / KB)       // 64 K-blocks
#define APAD 36      // padded A row stride in floats: conflict-free frag reads
#define BSZ  (KB * BN)          // 512 floats per gate per buffer

__global__ __launch_bounds__(128)
void lstm_fused_wmma_f32(const float* __restrict__ x,
                         const float* __restrict__ h_t,
                         const float* __restrict__ c_t,
                         const float* __restrict__ W_f, const float* __restrict__ b_f,
                         const float* __restrict__ W_i, const float* __restrict__ b_i,
                         const float* __restrict__ W_c, const float* __restrict__ b_c,
                         const float* __restrict__ W_o, const float* __restrict__ b_o,
                         float* __restrict__ outH, float* __restrict__ outH2,
                         float* __restrict__ outC)
{
    // Ping-pong LDS: 2 * (128*36 + 4*32*16) * 4B = 53,248 B
    __shared__ __align__(16) float a_lds[2][BM * APAD];
    __shared__ __align__(16) float b_lds[2][4 * BSZ];

    const int t    = threadIdx.x;      // 0..127 (4 waves)
    const int lane = t & 31;
    const int wv   = t >> 5;           // 0..3 : wave owns rows [wv*32, wv*32+32)
    const int nIdx = lane & 15;        // N (and A-row) index within fragment
    const int hi   = lane >> 4;        // lane-half: selects K+2 / M+8

    const int rowBase = blockIdx.x * BM;   // M offset into B_DIM
    const int nBase   = blockIdx.y * BN;   // N offset into H_DIM (same for every gate)

    // acc[m][g]: m = M sub-tile (0/1), g = gate (f,i,c,o)
    v8f acc[2][4] = {{{}, {}, {}, {}}, {{}, {}, {}, {}}};

    v4f aR[8];   // staged A: 8 float4 per thread per K-block
    v4f bR[4];   // staged B: 1 float4 per gate per thread per K-block

    // ---- global -> registers for K-block starting at kb ----
    auto loadGlobal = [&](int kb) {
        const float* Asrc = (kb < H_DIM) ? h_t : x;   // hx = [h_t | x], no concat
        const int    kOff = (kb < H_DIM) ? kb : (kb - H_DIM);
        #pragma unroll
        for (int i = 0; i < 8; ++i) {
            int idx = t + i * 128;        // float4 index 0..1023
            int r   = idx >> 3;           // 0..127
            int c4  = idx & 7;            // 0..7
            aR[i] = *(const v4f*)(Asrc + (size_t)(rowBase + r) * H_DIM + kOff + c4 * 4);
        }
        #pragma unroll
        for (int g = 0; g < 4; ++g) {     // thread t loads one float4 per gate
            const float* Wg = (g == 0) ? W_f : (g == 1) ? W_i : (g == 2) ? W_c : W_o;
            bR[g] = *(const v4f*)(Wg + (size_t)(kb + (t >> 2)) * H_DIM + nBase + (t & 3) * 4);
        }
    };

    // ---- registers -> LDS buffer ----
    auto storeLds = [&](int buf) {
        #pragma unroll
        for (int i = 0; i < 8; ++i) {
            int idx = t + i * 128;
            int r   = idx >> 3;
            int c4  = idx & 7;
            *(v4f*)(&a_lds[buf][r * APAD + c4 * 4]) = aR[i];
        }
        #pragma unroll
        for (int g = 0; g < 4; ++g)
            *(v4f*)(&b_lds[buf][g * BSZ + (t >> 2) * BN + (t & 3) * 4]) = bR[g];
    };

    // ---- LDS -> fragment registers for one K-step (2 A frags + 4 B frags) ----
    auto loadFrags = [&](int buf, int kk, v2f& a0, v2f& a1, v2f bg[4]) {
        const float* aBase = &a_lds[buf][(wv * 32 + nIdx) * APAD + 2 * hi];
        a0 = *(const v2f*)(aBase + kk);
        a1 = *(const v2f*)(aBase + 16 * APAD + kk);
        const float* bBase = &b_lds[buf][0];
        int bo = (kk + 2 * hi) * BN + nIdx;
        #pragma unroll
        for (int g = 0; g < 4; ++g) {
            v2f v = { bBase[g * BSZ + bo], bBase[g * BSZ + bo + BN] };
            bg[g] = v;
        }
    };

    loadGlobal(0);
    storeLds(0);
    __syncthreads();

    for (int it = 0; it < NKB; ++it) {
        const int  cur     = it & 1;
        const bool hasNext = (it + 1 < NKB);

        if (hasNext) loadGlobal((it + 1) * KB);   // overlap HBM with WMMA below

        // warm L2 two K-blocks ahead (global_prefetch_b8)
        if (it + 2 < NKB) {
            int kb2 = (it + 2) * KB;
            const float* As2 = (kb2 < H_DIM) ? h_t : x;
            int          kO2 = (kb2 < H_DIM) ? kb2 : (kb2 - H_DIM);
            __builtin_prefetch(As2 + (size_t)(rowBase + t) * H_DIM + kO2, 0, 1);
            const int gp = t >> 5;
            const float* Wp = (gp == 0) ? W_f : (gp == 1) ? W_i : (gp == 2) ? W_c : W_o;
            __builtin_prefetch(Wp + (size_t)(kb2 + (t & 31)) * H_DIM + nBase, 0, 1);
        }

        // ---- software-pipelined K-steps: next step's 6 LDS loads issue before
        //      this step's 8 WMMAs, so dscnt waits never drain to zero ----
        v2f a0c, a1c, bgc[4];
        loadFrags(cur, 0, a0c, a1c, bgc);
        #pragma unroll
        for (int kk = 0; kk < KB; kk += 4) {
            v2f a0n, a1n, bgn[4];
            if (kk + 4 < KB) loadFrags(cur, kk + 4, a0n, a1n, bgn);
            #pragma unroll
            for (int g = 0; g < 4; ++g) {
                acc[0][g] = __builtin_amdgcn_wmma_f32_16x16x4_f32(false, a0c, false, bgc[g],
                                                                  (short)0, acc[0][g], false, false);
                acc[1][g] = __builtin_amdgcn_wmma_f32_16x16x4_f32(false, a1c, false, bgc[g],
                                                                  (short)0, acc[1][g], false, false);
            }
            if (kk + 4 < KB) {
                a0c = a0n; a1c = a1n;
                #pragma unroll
                for (int g = 0; g < 4; ++g) bgc[g] = bgn[g];
            }
        }

        if (hasNext) storeLds(cur ^ 1);   // fill the other buffer for next iteration
        __syncthreads();
    }

    // ---- fused LSTM epilogue: bias + activations + cell update ----
    const int col = nBase + nIdx;
    const float bF = b_f[col], bI = b_i[col], bC = b_c[col], bO = b_o[col];

    #pragma unroll
    for (int m = 0; m < 2; ++m) {
        #pragma unroll
        for (int j = 0; j < 8; ++j) {
            int    r    = wv * 32 + m * 16 + j + 8 * hi;    // C/D layout: VGPR j -> M=j / j+8
            size_t gidx = (size_t)(rowBase + r) * H_DIM + col;
            float fg = 1.0f / (1.0f + __expf(-(acc[m][0][j] + bF)));
            float ig = 1.0f / (1.0f + __expf(-(acc[m][1][j] + bI)));
            float cg = tanhf(acc[m][2][j] + bC);
            float og = 1.0f / (1.0f + __expf(-(acc[m][3][j] + bO)));
            float cNew = fg * c_t[gidx] + ig * cg;
            float hNew = og * tanhf(cNew);
            outH [gidx] = hNew;
            outH2[gidx] = hNew;   // reference returns (h_new, (h_new, c_new))
            outC [gidx] = cNew;
        }
    }
}

extern "C" void kernel_launch(void* const* d_in, const int* in_sizes, int n_in,
                              void* d_out, int out_size, void* d_ws, size_t ws_size,
                              hipStream_t stream) {
    (void)in_sizes; (void)n_in; (void)out_size; (void)d_ws; (void)ws_size;
    const float* x   = (const float*)d_in[0];
    const float* h_t = (const float*)d_in[1];
    const float* c_t = (const float*)d_in[2];
    const float* W_f = (const float*)d_in[3];
    const float* b_f = (const float*)d_in[4];
    const float* W_i = (const float*)d_in[5];
    const float* b_i = (const float*)d_in[6];
    const float* W_c = (const float*)d_in[7];
    const float* b_c = (const float*)d_in[8];
    const float* W_o = (const float*)d_in[9];
    const float* b_o = (const float*)d_in[10];

    float* out   = (float*)d_out;
    float* outH  = out;
    float* outH2 = out + (size_t)B_DIM * H_DIM;
    float* outC  = out + 2 * (size_t)B_DIM * H_DIM;

    dim3 grid(B_DIM / BM, H_DIM / BN);   // (64, 64)
    dim3 block(128);                      // 4 waves
    hipLaunchKernelGGL(lstm_fused_wmma_f32, grid, block, 0, stream,
                       x, h_t, c_t, W_f, b_f, W_i, b_i, W_c, b_c, W_o, b_o,
                       outH, outH2, outC);
}